// Solov2Loss_39350490366643
// MI455X (gfx1250) — compile-verified
//
#include <hip/hip_runtime.h>

typedef __attribute__((ext_vector_type(2))) float v2f;
typedef __attribute__((ext_vector_type(8))) float v8f;

#define B_  2
#define E_  256
#define H_  192
#define W_  192
#define HW_ (H_*W_)
#define P_  16
#define C_  80
#define L_  5

// ws layout (floats): [0 .. 480): dice a,b,c per row  ((l*B+b)*P+p)*3 + {0,1,2}
//                     [480]: focal numerator  [481]: pos count
#define WS_FOCAL 480
#define WS_POS   481
#define WS_TOTAL 482

struct LevelPtrs {
    const float* kern[L_];
    const int*   pos[L_];
    int          g2[L_];
};

__global__ void init_ws_kernel(float* __restrict__ ws) {
    int i = threadIdx.x;
    if (i < WS_TOTAL) ws[i] = 0.0f;
}

__device__ __forceinline__ float fast_sigmoid(float x) {
    // v_exp_f32 + v_rcp_f32 (1-ulp), avoids the IEEE div_scale/div_fmas chain
    return __builtin_amdgcn_rcpf(1.0f + __expf(-x));
}

// Fused: dynamic-conv GEMM (WMMA f32 16x16x4) + sigmoid + dice partial sums.
// grid = (48, L*B), block = 256 (8 waves). 2304 N-tiles per (l,b) -> 6 tiles/wave.
__global__ void gemm_dice_kernel(const float* __restrict__ mask_out,
                                 const float* __restrict__ mask_targets,
                                 LevelPtrs lp,
                                 float* __restrict__ ws)
{
    __shared__ float Alds[P_ * E_];   // 16 KB: gathered positive kernels (A matrix)

    const int lb = blockIdx.y;        // l*B + b
    const int l  = lb / B_;
    const int b  = lb % B_;
    const int tid = threadIdx.x;

    const int    g2   = lp.g2[l];
    const float* kern = lp.kern[l];
    const int*   pos  = lp.pos[l];

    // Gather A[m][k] = kernel_flat[b][pos[b,m]][k] into LDS (coalesced per row).
    for (int idx = tid; idx < P_ * E_; idx += blockDim.x) {
        const int m = idx >> 8;           // E_ == 256
        const int k = idx & (E_ - 1);
        const int pj = pos[b * P_ + m];
        Alds[idx] = kern[((size_t)b * g2 + pj) * E_ + k];
    }
    __syncthreads();

    const int wave  = tid >> 5;
    const int lane  = tid & 31;
    const int laneM = lane & 15;          // column n within tile / row m of A
    const int hi    = lane >> 4;          // lane half (selects K pair / M half)

    const int nTiles     = HW_ / 16;                       // 2304
    const int wavesTotal = gridDim.x * (blockDim.x >> 5);  // 384
    const int waveId     = blockIdx.x * (blockDim.x >> 5) + wave;

    float da[8], db[8], dc[8];
#pragma unroll
    for (int j = 0; j < 8; ++j) { da[j] = 0.f; db[j] = 0.f; dc[j] = 0.f; }

    const float* Bbase = mask_out     + (size_t)b * E_ * HW_;
    const float* Tbase = mask_targets + ((size_t)b * L_ + l) * P_ * HW_;

    for (int t = waveId; t < nTiles; t += wavesTotal) {
        const int n = t * 16 + laneM;
        const float* Bp = Bbase + n;
        const float* Tp = Tbase + n;

        // Prefetch the 16 target rows this tile's epilogue will read
        // (global_prefetch_b8: counter-free, lands in cache ~64 iters ahead).
#pragma unroll
        for (int j = 0; j < 8; ++j) {
            __builtin_prefetch(Tp + (size_t)(j + 8 * hi) * HW_, 0, 1);
        }

        v8f acc = {0.f, 0.f, 0.f, 0.f, 0.f, 0.f, 0.f, 0.f};

#pragma unroll 8
        for (int k0 = 0; k0 < E_; k0 += 4) {
            const int kk = k0 + 2 * hi;
            // A 16x4 layout: v0 = A[m][k0 | k0+2], v1 = A[m][k0+1 | k0+3] -> b64 LDS read
            v2f av = *(const v2f*)&Alds[laneM * E_ + kk];
            // B 4x16 layout: v0 = B[k0 | k0+2][n], v1 = B[k0+1 | k0+3][n]
            v2f bv;
            bv.x = Bp[(size_t)kk * HW_];
            bv.y = Bp[(size_t)(kk + 1) * HW_];
            acc = __builtin_amdgcn_wmma_f32_16x16x4_f32(
                      false, av, false, bv, (short)0, acc, false, false);
        }

        // D layout: VGPR j, lane -> row p = j + 8*hi, col n = laneM (this tile)
#pragma unroll
        for (int j = 0; j < 8; ++j) {
            const float x  = acc[j];
            const float mp = fast_sigmoid(x);
            const int   p  = j + 8 * hi;
            const float mt = Tp[(size_t)p * HW_];
            da[j] += mp * mt;
            db[j] += mp * mp;
            dc[j] += mt * mt;
        }
    }

    // Reduce each quantity over the 16 lanes of each half (xor 1,2,4,8 stays in-half),
    // then one lane per half commits per-row atomics.
#pragma unroll
    for (int j = 0; j < 8; ++j) {
        float a = da[j], bb = db[j], c = dc[j];
        for (int m = 1; m < 16; m <<= 1) {
            a  += __shfl_xor(a,  m, 32);
            bb += __shfl_xor(bb, m, 32);
            c  += __shfl_xor(c,  m, 32);
        }
        if (laneM == 0) {
            const int p = j + 8 * hi;
            float* base = ws + ((size_t)lb * P_ + p) * 3;
            atomicAdd(base + 0, a);
            atomicAdd(base + 1, bb);
            atomicAdd(base + 2, c);
        }
    }
}

// Focal loss partials over one level's [B*g*g, C] grid.
__global__ void focal_kernel(const float* __restrict__ cate,
                             const int*   __restrict__ cate_t,
                             int ncells,
                             float* __restrict__ ws)
{
    __shared__ float sv[256];
    __shared__ float sc[256];
    const int tid = threadIdx.x;
    const int gid = blockIdx.x * blockDim.x + tid;

    float val = 0.f, cnt = 0.f;
    if (gid < ncells * C_) {
        const int cell = gid / C_;
        const int cls  = gid - cell * C_;
        const int t    = cate_t[cell];
        if (t > -1) {
            float p = cate[gid];
            p = fminf(fmaxf(p, 1e-4f), 1.0f - 1e-4f);
            const bool  gt = (cls == t);
            const float af = gt ? 0.25f : 0.75f;
            const float pt = gt ? p : (1.0f - p);
            const float om = 1.0f - pt;
            val = af * om * om * (-__logf(pt));
            if (cls == 0) cnt = 1.0f;
        }
    }
    sv[tid] = val;
    sc[tid] = cnt;
    __syncthreads();
    for (int s = 128; s > 0; s >>= 1) {
        if (tid < s) { sv[tid] += sv[tid + s]; sc[tid] += sc[tid + s]; }
        __syncthreads();
    }
    if (tid == 0) {
        if (sv[0] != 0.f) atomicAdd(ws + WS_FOCAL, sv[0]);
        if (sc[0] != 0.f) atomicAdd(ws + WS_POS,   sc[0]);
    }
}

__global__ void finalize_kernel(const float* __restrict__ ws,
                                float* __restrict__ out)
{
    __shared__ float red[256];
    const int tid = threadIdx.x;
    float v = 0.f;
    if (tid < L_ * B_ * P_) {
        const float* r = ws + (size_t)tid * 3;
        v = 1.0f - 2.0f * r[0] / (r[1] + r[2] + 1e-4f);
    }
    red[tid] = v;
    __syncthreads();
    for (int s = 128; s > 0; s >>= 1) {
        if (tid < s) red[tid] += red[tid + s];
        __syncthreads();
    }
    if (tid == 0) {
        out[0] = 3.0f * red[0] / (float)(L_ * B_ * P_);   // MASK_W * mask_loss
        out[1] = 1.0f * ws[WS_FOCAL] / ws[WS_POS];        // CLS_W  * cls_loss
    }
}

extern "C" void kernel_launch(void* const* d_in, const int* in_sizes, int n_in,
                              void* d_out, int out_size, void* d_ws, size_t ws_size,
                              hipStream_t stream)
{
    (void)in_sizes; (void)n_in; (void)out_size; (void)ws_size;

    // setup_inputs() dict order:
    // 0: mask_out; then per level i: 1+4i kernel, 2+4i cate, 3+4i pos, 4+4i cate_t; 21: mask_targets
    const float* mask_out     = (const float*)d_in[0];
    const float* mask_targets = (const float*)d_in[21];

    static const int GRIDS[L_] = {40, 36, 24, 16, 12};

    LevelPtrs lp;
    const float* cate[L_];
    const int*   cate_t[L_];
    for (int i = 0; i < L_; ++i) {
        lp.kern[i] = (const float*)d_in[1 + 4 * i];
        cate[i]    = (const float*)d_in[2 + 4 * i];
        lp.pos[i]  = (const int*)  d_in[3 + 4 * i];
        cate_t[i]  = (const int*)  d_in[4 + 4 * i];
        lp.g2[i]   = GRIDS[i] * GRIDS[i];
    }

    float* ws  = (float*)d_ws;
    float* out = (float*)d_out;

    init_ws_kernel<<<1, 512, 0, stream>>>(ws);

    gemm_dice_kernel<<<dim3(48, L_ * B_), 256, 0, stream>>>(mask_out, mask_targets, lp, ws);

    for (int i = 0; i < L_; ++i) {
        const int ncells = B_ * lp.g2[i];
        const int total  = ncells * C_;
        focal_kernel<<<(total + 255) / 256, 256, 0, stream>>>(cate[i], cate_t[i], ncells, ws);
    }

    finalize_kernel<<<1, 256, 0, stream>>>(ws, out);
}